// FusedAttentionBias_4080218931663
// MI455X (gfx1250) — compile-verified
//
#include <hip/hip_runtime.h>
#include <hip/hip_bf16.h>

// Problem constants (B=2, N=S=2048, H=16, E=64), fp32 throughout.
#define BB 2
#define NN 2048
#define SS 2048
#define HH 16
#define EE 64
#define SCALE 0.125f

typedef float v2f __attribute__((ext_vector_type(2)));
typedef float v8f __attribute__((ext_vector_type(8)));

// ---------------------------------------------------------------------------
// Kernel 1: scores = K·Q^T * SCALE + bias, causal mask (n>s -> -inf),
//           softmax over n, write attn_weights [B,H,S,N].
// One workgroup = 16 key rows (s) x full n strip (2048), strip lives in LDS.
// ---------------------------------------------------------------------------
#define K1_LDS_STRIDE 2052   // 2048 + 4 pad: kills 2-way bank conflict

__global__ __launch_bounds__(128) void k1_scores_softmax(
    const float* __restrict__ q,     // [B,N,H,E]
    const float* __restrict__ k,     // [B,S,H,E]
    const float* __restrict__ bias,  // [1,H,S,N]
    float* __restrict__ w)           // [B,H,S,N]
{
  __shared__ float sc[16 * K1_LDS_STRIDE];  // 16 rows x 2048 scores (padded)
  __shared__ float red[128];                // [0..63] row maxes, [64..127] row sums (x4 waves)

  const int tid  = threadIdx.x;
  const int wid  = tid >> 5;
  const int lane = tid & 31;
  const int m    = lane & 15;   // row (A) / col (B,D) index within tile
  const int hi   = lane >> 4;   // lane-half: selects K=0,1 vs K=2,3 of the x4 frag

  const int s0 = blockIdx.x * 16;  // key-row block
  const int h  = blockIdx.y;
  const int b  = blockIdx.z;

  // --- Phase 1: WMMA scores into LDS -------------------------------------
  // A = K rows s0..s0+15 (16x4 f32 frag per 4-wide e chunk): per-lane float2
  //   a.x = K[s0+m][e0+2*hi], a.y = K[s0+m][e0+2*hi+1]
  v2f afrag[16];
  const float* kbase = k + ((size_t)(b * SS + s0 + m) * HH + h) * EE;
#pragma unroll
  for (int e4 = 0; e4 < 16; ++e4)
    afrag[e4] = *(const v2f*)(kbase + e4 * 4 + 2 * hi);

  for (int t = wid; t < NN / 16; t += 4) {   // uniform per wave: EXEC stays full
    const int n0 = t * 16;
    const float* qbase = q + ((size_t)(b * NN + n0 + m) * HH + h) * EE;
    v2f bfrag[16];
#pragma unroll
    for (int e4 = 0; e4 < 16; ++e4)
      bfrag[e4] = *(const v2f*)(qbase + e4 * 4 + 2 * hi);

    v8f c = {};
#pragma unroll
    for (int e4 = 0; e4 < 16; ++e4)
      c = __builtin_amdgcn_wmma_f32_16x16x4_f32(
              false, afrag[e4], false, bfrag[e4], (short)0, c, false, false);

    // D layout: VGPR v, lane -> row (v + 8*hi), col m
#pragma unroll
    for (int v = 0; v < 8; ++v)
      sc[(v + 8 * hi) * K1_LDS_STRIDE + n0 + m] = c[v] * SCALE;
  }
  __syncthreads();

  // --- Phase 2a: bias + causal mask + row max ----------------------------
  // bias rows for s0..s0+15 are contiguous: fully coalesced streaming read.
  const float* bbase = bias + ((size_t)h * SS + s0) * NN;
#pragma unroll 1
  for (int r = 0; r < 16; ++r) {
    const int s = s0 + r;
    float mx = -__builtin_inff();
#pragma unroll
    for (int j = 0; j < 16; ++j) {
      const int col = tid + 128 * j;
      float val = sc[r * K1_LDS_STRIDE + col] +
                  __builtin_nontemporal_load(bbase + (size_t)r * NN + col);
      if (col > s) val = -__builtin_inff();
      sc[r * K1_LDS_STRIDE + col] = val;
      mx = fmaxf(mx, val);
    }
    for (int off = 16; off > 0; off >>= 1)
      mx = fmaxf(mx, __shfl_xor(mx, off, 32));
    if (lane == 0) red[r * 4 + wid] = mx;
  }
  __syncthreads();

  // --- Phase 2b: exp + row sum -------------------------------------------
#pragma unroll 1
  for (int r = 0; r < 16; ++r) {
    const float rm = fmaxf(fmaxf(red[r * 4 + 0], red[r * 4 + 1]),
                           fmaxf(red[r * 4 + 2], red[r * 4 + 3]));
    float sum = 0.f;
#pragma unroll
    for (int j = 0; j < 16; ++j) {
      const int col = tid + 128 * j;
      const float e = __expf(sc[r * K1_LDS_STRIDE + col] - rm);  // exp(-inf)=0 handles mask
      sc[r * K1_LDS_STRIDE + col] = e;
      sum += e;
    }
    for (int off = 16; off > 0; off >>= 1)
      sum += __shfl_xor(sum, off, 32);
    if (lane == 0) red[64 + r * 4 + wid] = sum;
  }
  __syncthreads();

  // --- Phase 2c: normalize, NT-stream weights out ------------------------
  float* wbase = w + (((size_t)(b * HH + h) * SS + s0) * NN);
#pragma unroll 1
  for (int r = 0; r < 16; ++r) {
    const float inv = 1.0f / (red[64 + r * 4 + 0] + red[64 + r * 4 + 1] +
                              red[64 + r * 4 + 2] + red[64 + r * 4 + 3]);
#pragma unroll
    for (int j = 0; j < 16; ++j) {
      const int col = tid + 128 * j;
      __builtin_nontemporal_store(sc[r * K1_LDS_STRIDE + col] * inv,
                                  wbase + (size_t)r * NN + col);
    }
  }
}

// ---------------------------------------------------------------------------
// Kernel 2: out[b,n,h,e] = sum_s w[b,h,s,n] * V[b,s,h,e]   (= W^T · V)
// WG = 256 threads (8 waves), owns 256 n-rows; V staged in LDS in 128-row
// chunks (32x reuse). A = W^T frags NT-loaded from global (stream-once).
// ---------------------------------------------------------------------------
#define K2_SC 128          // s-chunk staged in LDS
#define K2_VSTR 68         // 64 + 4 pad floats per V row in LDS

__global__ __launch_bounds__(256) void k2_weighted_value(
    const float* __restrict__ w,   // [B,H,S,N]
    const float* __restrict__ v,   // [B,S,H,E]
    float* __restrict__ out)       // [B,N,H,E]
{
  __shared__ float vlds[K2_SC * K2_VSTR];  // 34.8 KB

  const int tid  = threadIdx.x;
  const int wid  = tid >> 5;
  const int lane = tid & 31;
  const int m    = lane & 15;
  const int hi   = lane >> 4;

  const int h = blockIdx.y;
  const int b = blockIdx.z;
  const int n0w = blockIdx.x * 256 + wid * 32;   // this wave's 32 n-rows

  const float* wslice = w + ((size_t)(b * HH + h) * SS) * NN;

  v8f acc[2][4];   // [n-tile][e-tile], e-tiles of 16 cover E=64
#pragma unroll
  for (int nt = 0; nt < 2; ++nt)
#pragma unroll
    for (int et = 0; et < 4; ++et)
      acc[nt][et] = (v8f){};

  for (int sbase = 0; sbase < SS; sbase += K2_SC) {
    // ---- stage V[sbase..+127][0..63] into LDS (coalesced float4) --------
#pragma unroll
    for (int i = 0; i < (K2_SC * EE / 4) / 256; ++i) {   // 2048 float4 / 256 thr
      const int f   = tid + 256 * i;
      const int row = f >> 4;          // 16 float4 per row
      const int e4  = (f & 15) * 4;
      const float4 val = *(const float4*)(
          v + ((size_t)(b * SS + sbase + row) * HH + h) * EE + e4);
      *(float4*)(&vlds[row * K2_VSTR + e4]) = val;
    }
    __syncthreads();

    // ---- GEMM over this chunk: 32 k-steps of 4 --------------------------
    for (int kk = 0; kk < K2_SC / 4; ++kk) {
      const int sk = sbase + kk * 4;
      // A frags: a.x = W[sk+2*hi][n0+m], a.y = W[sk+2*hi+1][n0+m]
      v2f a[2];
#pragma unroll
      for (int nt = 0; nt < 2; ++nt) {
        const float* ap = wslice + (size_t)(sk + 2 * hi) * NN + (n0w + nt * 16 + m);
        a[nt].x = __builtin_nontemporal_load(ap);
        a[nt].y = __builtin_nontemporal_load(ap + NN);
      }
      // B frags from LDS: b.x = V[sk+2*hi][e0+m], b.y = V[sk+2*hi+1][e0+m]
#pragma unroll
      for (int et = 0; et < 4; ++et) {
        const int e0 = et * 16;
        v2f bf;
        bf.x = vlds[(kk * 4 + 2 * hi) * K2_VSTR + e0 + m];
        bf.y = vlds[(kk * 4 + 2 * hi + 1) * K2_VSTR + e0 + m];
#pragma unroll
        for (int nt = 0; nt < 2; ++nt)
          acc[nt][et] = __builtin_amdgcn_wmma_f32_16x16x4_f32(
              false, a[nt], false, bf, (short)0, acc[nt][et], false, false);
      }
    }
    __syncthreads();
  }

  // ---- epilogue: D[v][lane] -> out[b, n0+v+8*hi, h, e0+m] ----------------
#pragma unroll
  for (int nt = 0; nt < 2; ++nt) {
#pragma unroll
    for (int et = 0; et < 4; ++et) {
#pragma unroll
      for (int vv = 0; vv < 8; ++vv) {
        const int n = n0w + nt * 16 + vv + 8 * hi;
        __builtin_nontemporal_store(
            acc[nt][et][vv],
            out + ((size_t)(b * NN + n) * HH + h) * EE + et * 16 + m);
      }
    }
  }
}

// ---------------------------------------------------------------------------
extern "C" void kernel_launch(void* const* d_in, const int* in_sizes, int n_in,
                              void* d_out, int out_size, void* d_ws, size_t ws_size,
                              hipStream_t stream) {
  const float* q    = (const float*)d_in[0];   // [B,N,H,E]
  const float* k    = (const float*)d_in[1];   // [B,S,H,E]
  const float* v    = (const float*)d_in[2];   // [B,S,H,E]
  const float* bias = (const float*)d_in[3];   // [1,H,S,N]

  float* out = (float*)d_out;                             // [B,N,H,E]
  float* w   = out + (size_t)BB * NN * HH * EE;           // [B,H,S,N]

  dim3 g1(SS / 16, HH, BB);
  k1_scores_softmax<<<g1, 128, 0, stream>>>(q, k, bias, w);

  dim3 g2(NN / 256, HH, BB);
  k2_weighted_value<<<g2, 256, 0, stream>>>(w, v, out);
}